// GCN_77421080478261
// MI455X (gfx1250) — compile-verified
//
#include <hip/hip_runtime.h>
#include <hip/hip_bf16.h>

#define N_NODES 100000
#define N_EDGES 3200000
#define N_GRAPHS 64
#define HID 64
#define BN_EPS 1e-5f

typedef __attribute__((ext_vector_type(2))) float v2f;
typedef __attribute__((ext_vector_type(8))) float v8f;

// ---------------------------------------------------------------------------
// BatchNorm(4) statistics: per-feature sum and sum-of-squares over all nodes.
// ---------------------------------------------------------------------------
__global__ void bn0_stats_kernel(const float* __restrict__ x, float* __restrict__ acc, int n) {
  __shared__ float red[256];
  float s[8] = {0.f, 0.f, 0.f, 0.f, 0.f, 0.f, 0.f, 0.f};
  for (int i = blockIdx.x * blockDim.x + threadIdx.x; i < n; i += gridDim.x * blockDim.x) {
    float4 v = ((const float4*)x)[i];
    s[0] += v.x; s[1] += v.y; s[2] += v.z; s[3] += v.w;
    s[4] += v.x * v.x; s[5] += v.y * v.y; s[6] += v.z * v.z; s[7] += v.w * v.w;
  }
  for (int q = 0; q < 8; ++q) {
    red[threadIdx.x] = s[q];
    __syncthreads();
    for (int off = 128; off > 0; off >>= 1) {
      if (threadIdx.x < (unsigned)off) red[threadIdx.x] += red[threadIdx.x + off];
      __syncthreads();
    }
    if (threadIdx.x == 0) atomicAdd(&acc[q], red[0]);
    __syncthreads();
  }
}

// ---------------------------------------------------------------------------
// Fold BN0 into W0:  bn(x)@W0 = x @ (scale_k * W0[k,:]) + (shift @ W0)
// ---------------------------------------------------------------------------
__global__ void prep_kernel(const float* __restrict__ acc,
                            const float* __restrict__ bn_g, const float* __restrict__ bn_b,
                            const float* __restrict__ W0,
                            float* __restrict__ W0p, float* __restrict__ bias0p,
                            float inv_n) {
  __shared__ float scale[4], shift[4];
  int t = threadIdx.x;
  if (t < 4) {
    float mu  = acc[t] * inv_n;
    float var = acc[4 + t] * inv_n - mu * mu;
    float sc  = bn_g[t] * rsqrtf(var + BN_EPS);
    scale[t] = sc;
    shift[t] = bn_b[t] - mu * sc;
  }
  __syncthreads();
  if (t < 64) {
    float b = 0.f;
    #pragma unroll
    for (int k = 0; k < 4; ++k) {
      float w = W0[k * 64 + t];
      W0p[k * 64 + t] = scale[k] * w;
      b += shift[k] * w;
    }
    bias0p[t] = b;
  }
}

// ---------------------------------------------------------------------------
// Degree: deg[dst] += 1 over edges (self-loop +1 added in finalize).
// ---------------------------------------------------------------------------
__global__ void deg_count_kernel(const long long* __restrict__ ei, float* __restrict__ deg, int E) {
  int e = blockIdx.x * blockDim.x + threadIdx.x;
  if (e < E) atomicAdd(&deg[(int)ei[(size_t)E + e]], 1.0f);
}

__global__ void deg_finalize_kernel(const float* __restrict__ deg,
                                    float* __restrict__ dinv, float* __restrict__ invdeg, int n) {
  int i = blockIdx.x * blockDim.x + threadIdx.x;
  if (i < n) {
    float d = deg[i] + 1.0f;
    dinv[i] = rsqrtf(d);
    invdeg[i] = 1.0f / d;
  }
}

// ---------------------------------------------------------------------------
// Layer 1 GEMM: H[N,64] = x[N,4] @ W0p[4,64] + bias0p.   One f32 WMMA / tile.
// Block = 128 threads (4 waves); wave w -> cols [16w,16w+16), block -> 16 rows.
// ---------------------------------------------------------------------------
__global__ void gemm4_wmma_kernel(const float* __restrict__ x,
                                  const float* __restrict__ W0p,
                                  const float* __restrict__ bias0p,
                                  float* __restrict__ Hout) {
  int tid = threadIdx.x;
  int wave = tid >> 5, lane = tid & 31;
  int rowBase = blockIdx.x * 16;
  int colBase = wave * 16;
  int m = lane & 15;
  int khalf = (lane >> 4) * 2;   // lanes 0-15 -> K=0,1 ; lanes 16-31 -> K=2,3

  const float* xr = x + (size_t)(rowBase + m) * 4;
  v2f a, b;
  a.x = xr[khalf];
  a.y = xr[khalf + 1];
  b.x = W0p[khalf * 64 + colBase + m];
  b.y = W0p[(khalf + 1) * 64 + colBase + m];

  v8f c = {};
  c = __builtin_amdgcn_wmma_f32_16x16x4_f32(false, a, false, b, (short)0, c, false, false);

  float bias = bias0p[colBase + m];
  int rOff = (lane >> 4) * 8;    // lanes 16-31 hold rows 8..15
  float* out = Hout + (size_t)rowBase * 64 + colBase + m;
  #pragma unroll
  for (int r = 0; r < 8; ++r) out[(size_t)(r + rOff) * 64] = c[r] + bias;
}

// ---------------------------------------------------------------------------
// Layer 2 GEMM: Hout[N,64] = Hin[N,64] @ W[64,64].  16 accumulating f32 WMMAs.
// W (16KB) and the 16x64 A-tile staged in LDS, shared by all 4 waves.
// ---------------------------------------------------------------------------
__global__ void gemm64_wmma_kernel(const float* __restrict__ Hin,
                                   const float* __restrict__ W,
                                   float* __restrict__ Hout) {
  __shared__ float sW[64 * 64];
  __shared__ float sA[16 * 64];
  int tid = threadIdx.x;

  const float4* W4 = (const float4*)W;
  float4* sW4 = (float4*)sW;
  for (int i = tid; i < 64 * 64 / 4; i += 128) sW4[i] = W4[i];

  int rowBase = blockIdx.x * 16;
  const float4* H4 = (const float4*)(Hin + (size_t)rowBase * 64);
  float4* sA4 = (float4*)sA;
  for (int i = tid; i < 16 * 64 / 4; i += 128) sA4[i] = H4[i];
  __syncthreads();

  int wave = tid >> 5, lane = tid & 31;
  int colBase = wave * 16;
  int m = lane & 15;
  int khalf = (lane >> 4) * 2;

  v8f c = {};
  #pragma unroll
  for (int k0 = 0; k0 < 64; k0 += 4) {
    v2f a, b;
    a.x = sA[m * 64 + k0 + khalf];
    a.y = sA[m * 64 + k0 + khalf + 1];
    b.x = sW[(k0 + khalf) * 64 + colBase + m];
    b.y = sW[(k0 + khalf + 1) * 64 + colBase + m];
    c = __builtin_amdgcn_wmma_f32_16x16x4_f32(false, a, false, b, (short)0, c, false, false);
  }

  int rOff = (lane >> 4) * 8;
  float* out = Hout + (size_t)rowBase * 64 + colBase + m;
  #pragma unroll
  for (int r = 0; r < 8; ++r) out[(size_t)(r + rOff) * 64] = c[r];
}

// ---------------------------------------------------------------------------
// Edge aggregation: s[dst,:] += h[src,:] * dinv[src].  One wave per edge,
// each lane handles 2 features (float2 load + 2 f32 atomics).
// ---------------------------------------------------------------------------
__global__ void edge_agg_kernel(const long long* __restrict__ ei,
                                const float* __restrict__ h,
                                const float* __restrict__ dinv,
                                float* __restrict__ s, int E) {
  int e = (int)((blockIdx.x * (size_t)blockDim.x + threadIdx.x) >> 5);
  int lane = threadIdx.x & 31;
  if (e >= E) return;
  int src = (int)ei[e];
  int dst = (int)ei[(size_t)E + e];
  float w = dinv[src];
  float2 v = ((const float2*)(h + (size_t)src * 64))[lane];
  float* sp = s + (size_t)dst * 64 + lane * 2;
  atomicAdd(sp, v.x * w);
  atomicAdd(sp + 1, v.y * w);
}

// ---------------------------------------------------------------------------
// GCN finalize: out = s * dinv[i] + h * invdeg[i] + bias[f]
// ---------------------------------------------------------------------------
__global__ void gcn_finalize_kernel(const float* __restrict__ s,
                                    const float* __restrict__ h,
                                    const float* __restrict__ dinv,
                                    const float* __restrict__ invdeg,
                                    const float* __restrict__ bias,
                                    float* __restrict__ out, int n) {
  int idx = blockIdx.x * blockDim.x + threadIdx.x;
  if (idx >= n * 64) return;
  int i = idx >> 6, f = idx & 63;
  out[idx] = s[idx] * dinv[i] + h[idx] * invdeg[i] + bias[f];
}

// ---------------------------------------------------------------------------
// Global add pool: g[batch[i], f] += h[i, f]
// ---------------------------------------------------------------------------
__global__ void pool_kernel(const float* __restrict__ h,
                            const long long* __restrict__ batch,
                            float* __restrict__ g, int n) {
  int idx = blockIdx.x * blockDim.x + threadIdx.x;
  if (idx >= n * 64) return;
  int i = idx >> 6, f = idx & 63;
  atomicAdd(&g[(int)batch[i] * 64 + f], h[idx]);
}

// ---------------------------------------------------------------------------
// Head: BN over graphs + Linear(64,64) + Linear(64,64) + Linear(64,1).
// One block, 64 threads (thread = feature for BN, thread = graph for MLP).
// ---------------------------------------------------------------------------
__global__ void head_kernel(const float* __restrict__ g,
                            const float* __restrict__ bn_g, const float* __restrict__ bn_b,
                            const float* __restrict__ l0W, const float* __restrict__ l0b,
                            const float* __restrict__ l1W, const float* __restrict__ l1b,
                            const float* __restrict__ oW, const float* __restrict__ ob,
                            float* __restrict__ out) {
  __shared__ float gn[64 * 64];
  __shared__ float y0[64 * 64];
  int t = threadIdx.x;  // 0..63

  // BN over the graph axis, one feature per thread
  float mu = 0.f;
  for (int i = 0; i < 64; ++i) mu += g[i * 64 + t];
  mu *= (1.0f / 64.0f);
  float var = 0.f;
  for (int i = 0; i < 64; ++i) { float d = g[i * 64 + t] - mu; var += d * d; }
  var *= (1.0f / 64.0f);
  float sc = bn_g[t] * rsqrtf(var + BN_EPS);
  float sh = bn_b[t] - mu * sc;
  for (int i = 0; i < 64; ++i) gn[i * 64 + t] = g[i * 64 + t] * sc + sh;
  __syncthreads();

  // MLP: one graph per thread
  for (int f = 0; f < 64; ++f) {
    float acc = l0b[f];
    for (int k = 0; k < 64; ++k) acc += gn[t * 64 + k] * l0W[k * 64 + f];
    y0[t * 64 + f] = acc;
  }
  __syncthreads();
  float y1[64];
  for (int f = 0; f < 64; ++f) {
    float acc = l1b[f];
    for (int k = 0; k < 64; ++k) acc += y0[t * 64 + k] * l1W[k * 64 + f];
    y1[f] = acc;
  }
  float o = ob[0];
  for (int k = 0; k < 64; ++k) o += y1[k] * oW[k];
  out[t] = o;
}

// ---------------------------------------------------------------------------
extern "C" void kernel_launch(void* const* d_in, const int* in_sizes, int n_in,
                              void* d_out, int out_size, void* d_ws, size_t ws_size,
                              hipStream_t stream) {
  (void)in_sizes; (void)n_in; (void)out_size; (void)ws_size;

  const float*     x      = (const float*)d_in[0];
  const long long* ei     = (const long long*)d_in[1];   // [2, E] int64
  const long long* batch  = (const long long*)d_in[2];   // [N]   int64
  const float*     bn0_g  = (const float*)d_in[3];
  const float*     bn0_b  = (const float*)d_in[4];
  const float*     W0     = (const float*)d_in[5];
  const float*     b0     = (const float*)d_in[6];
  const float*     W1     = (const float*)d_in[7];
  const float*     b1     = (const float*)d_in[8];
  const float*     bn1_g  = (const float*)d_in[9];
  const float*     bn1_b  = (const float*)d_in[10];
  const float*     l0W    = (const float*)d_in[11];
  const float*     l0b    = (const float*)d_in[12];
  const float*     l1W    = (const float*)d_in[13];
  const float*     l1b    = (const float*)d_in[14];
  const float*     oW     = (const float*)d_in[15];
  const float*     ob     = (const float*)d_in[16];
  float*           out    = (float*)d_out;

  const int N = N_NODES, E = N_EDGES;

  // Workspace layout (floats)
  float* A      = (float*)d_ws;                 // N*64 : h0 -> out0 -> s1 -> out? (ping)
  float* B      = A + (size_t)N * 64;           // N*64 : s0 -> h1 -> out1       (pong)
  float* deg    = B + (size_t)N * 64;           // N
  float* dinv   = deg + N;                      // N
  float* invdeg = dinv + N;                     // N
  float* acc    = invdeg + N;                   // 8
  float* W0p    = acc + 8;                      // 4*64
  float* bias0p = W0p + 256;                    // 64
  float* gbuf   = bias0p + 64;                  // 64*64

  // Zero accumulators (graph-capture safe)
  hipMemsetAsync(acc,  0, 8 * sizeof(float), stream);
  hipMemsetAsync(deg,  0, (size_t)N * sizeof(float), stream);
  hipMemsetAsync(gbuf, 0, 64 * 64 * sizeof(float), stream);

  // BN0 stats + fold into W0
  bn0_stats_kernel<<<512, 256, 0, stream>>>(x, acc, N);
  prep_kernel<<<1, 64, 0, stream>>>(acc, bn0_g, bn0_b, W0, W0p, bias0p, 1.0f / (float)N);

  // Degrees
  deg_count_kernel<<<(E + 255) / 256, 256, 0, stream>>>(ei, deg, E);
  deg_finalize_kernel<<<(N + 255) / 256, 256, 0, stream>>>(deg, dinv, invdeg, N);

  // Layer 1: h0 = bn(x) @ W0  (WMMA, K=4)   -> A
  gemm4_wmma_kernel<<<N / 16, 128, 0, stream>>>(x, W0p, bias0p, A);

  // agg1 -> B ; out0 = s*dinv + h0*invdeg + b0 -> A (in place over h0)
  hipMemsetAsync(B, 0, (size_t)N * 64 * sizeof(float), stream);
  edge_agg_kernel<<<(E + 7) / 8, 256, 0, stream>>>(ei, A, dinv, B, E);
  gcn_finalize_kernel<<<(N * 64 + 255) / 256, 256, 0, stream>>>(B, A, dinv, invdeg, b0, A, N);

  // Layer 2: h1 = out0 @ W1 (WMMA, K=64)    -> B
  gemm64_wmma_kernel<<<N / 16, 128, 0, stream>>>(A, W1, B);

  // agg2 -> A ; out1 = s*dinv + h1*invdeg + b1 -> B (in place over h1)
  hipMemsetAsync(A, 0, (size_t)N * 64 * sizeof(float), stream);
  edge_agg_kernel<<<(E + 7) / 8, 256, 0, stream>>>(ei, B, dinv, A, E);
  gcn_finalize_kernel<<<(N * 64 + 255) / 256, 256, 0, stream>>>(A, B, dinv, invdeg, b1, B, N);

  // Pool + head
  pool_kernel<<<(N * 64 + 255) / 256, 256, 0, stream>>>(B, batch, gbuf, N);
  head_kernel<<<1, 64, 0, stream>>>(gbuf, bn1_g, bn1_b, l0W, l0b, l1W, l1b, oW, ob, out);
}